// MultiHeadDampedEMA_30322469110541
// MI455X (gfx1250) — compile-verified
//
#include <hip/hip_runtime.h>

#define DCH     2048          // NUM_HEADS * HEAD_DIM
#define BATCH   4
#define TLEN    4096
#define SEGS    16
#define SEGLEN  (TLEN / SEGS) // 256
#define NTILE   (DCH / 16)    // 128 channel tiles
#define EPS_A   1e-6f

typedef __attribute__((ext_vector_type(2))) float v2f;
typedef __attribute__((ext_vector_type(8))) float v8f;

__device__ __forceinline__ float get_alpha(const float* __restrict__ alpha_logit) {
    float a = 1.0f / (1.0f + __expf(-alpha_logit[0]));
    return fminf(fmaxf(a, EPS_A), 1.0f - EPS_A);
}

// Pass 1: per (b, segment, channel) run local EMA with zero init; emit final value.
// Regular-temporal loads deliberately pull all of x (134 MB) into the 192 MB L2
// so pass 3's re-read hits in-cache.
__global__ void ema_seg_final(const float* __restrict__ x,
                              const float* __restrict__ alpha_logit,
                              float* __restrict__ segf) {
    const float a = get_alpha(alpha_logit);
    const float c = 1.0f - a;
    int gid = blockIdx.x * blockDim.x + threadIdx.x;     // ((b*SEGS + s)*DCH + d)
    int d  = gid % DCH;
    int bs = gid / DCH;
    int s  = bs % SEGS;
    int b  = bs / SEGS;
    const float* xp = x + ((size_t)(b * TLEN + s * SEGLEN)) * DCH + d;
    float h = 0.0f;
#pragma unroll 8
    for (int i = 0; i < SEGLEN; ++i) {
        h = fmaf(c, h, a * xp[(size_t)i * DCH]);
    }
    segf[gid] = h;
}

// Pass 2: exclusive carry across segments: C[0]=0; C[s] = f[s-1] + c^SEGLEN * C[s-1].
__global__ void ema_carry_scan(const float* __restrict__ segf,
                               const float* __restrict__ alpha_logit,
                               float* __restrict__ carry) {
    const float a = get_alpha(alpha_logit);
    const float c = 1.0f - a;
    float cL = c;                           // c^(2^8) = c^256 = c^SEGLEN
#pragma unroll
    for (int i = 0; i < 8; ++i) cL *= cL;
    int gid = blockIdx.x * blockDim.x + threadIdx.x;     // b*DCH + d
    int d = gid % DCH;
    int b = gid / DCH;
    float C = 0.0f;
    for (int s = 0; s < SEGS; ++s) {
        size_t o = ((size_t)(b * SEGS + s)) * DCH + d;
        carry[o] = C;
        C = fmaf(cL, C, segf[o]);
    }
}

// Pass 3: one wave per (b, segment, 16-channel tile). Each 16-step chunk:
//   H(16x16) = K(16x16, lower-tri decay) @ X(16x16) + outer(c^{m+1}, carry)
// K@X done with 4 accumulating v_wmma_f32_16x16x4_f32; carry update via row-15
// ds_bpermute. Output stores are non-temporal so x stays resident in L2.
__global__ void ema_scan_wmma(const float* __restrict__ x,
                              const float* __restrict__ alpha_logit,
                              const float* __restrict__ carryIn,
                              float* __restrict__ out) {
    const float a = get_alpha(alpha_logit);
    const float c = 1.0f - a;

    const int lane   = threadIdx.x & 31;
    const int laneHi = lane >> 4;           // 0: lanes 0-15, 1: lanes 16-31
    const int lanLo  = lane & 15;

    int w    = blockIdx.x * (blockDim.x >> 5) + (threadIdx.x >> 5);
    int tile = w % NTILE;
    int bs   = w / NTILE;
    int s    = bs % SEGS;
    int b    = bs / SEGS;
    int d0   = tile * 16;

    // A operand (16x4 f32 layout): lane holds row m=lanLo; VGPR r holds
    // column j = 4*sk + r + 2*laneHi of the decay matrix K[m][j] = a*c^(m-j), j<=m.
    v2f A[4];
#pragma unroll
    for (int sk = 0; sk < 4; ++sk) {
#pragma unroll
        for (int r = 0; r < 2; ++r) {
            int j = 4 * sk + r + 2 * laneHi;
            int m = lanLo;
            float v = 0.0f;
            if (m >= j) {
                float p = a;
                for (int q = 0; q < m - j; ++q) p *= c;
                v = p;
            }
            A[sk][r] = v;
        }
    }

    // true h entering this segment, for channel d0+lanLo (all 32 lanes mirror 16 chans)
    float carry = carryIn[((size_t)(b * SEGS + s)) * DCH + d0 + lanLo];

    // c^{m+1} starting power for this lane's rows (m = r + 8*laneHi)
    float cpow0 = c;                        // c^1
    if (laneHi) {
        float c8 = c * c; c8 *= c8; c8 *= c8;   // c^8
        cpow0 = c8 * c;                     // c^9
    }

    const size_t rs = DCH;
    const float* xbase = x   + ((size_t)(b * TLEN + s * SEGLEN)) * rs;
    float*       obase = out + ((size_t)(b * TLEN + s * SEGLEN)) * rs;

    // Unroll x2: next chunk's loads + WMMAs are independent of this chunk's carry,
    // so the compiler can overlap them with the carry-add / bpermute / store tail.
#pragma unroll 2
    for (int ch = 0; ch < SEGLEN / 16; ++ch) {
        int t0 = ch * 16;
        // B operand (4x16 f32 layout): VGPR r, lane half selects K-row
        // time j = 4*sk + r + 2*laneHi, channel N = lanLo.
        v2f Bm[4];
#pragma unroll
        for (int sk = 0; sk < 4; ++sk) {
#pragma unroll
            for (int r = 0; r < 2; ++r) {
                int j = 4 * sk + r + 2 * laneHi;
                Bm[sk][r] = xbase[(size_t)(t0 + j) * rs + d0 + lanLo];
            }
        }

        v8f acc = {};
#pragma unroll
        for (int sk = 0; sk < 4; ++sk) {
            acc = __builtin_amdgcn_wmma_f32_16x16x4_f32(
                false, A[sk], false, Bm[sk], (short)0, acc, false, false);
        }

        // add cross-chunk carry: acc[r] (row m = r + 8*laneHi) += c^{m+1} * carry
        float pw = cpow0;
#pragma unroll
        for (int r = 0; r < 8; ++r) {
            acc[r] = fmaf(pw, carry, acc[r]);
            pw *= c;
        }

        // store tile (non-temporal: don't evict x from L2):
        // lane writes column d0+lanLo, rows m = r + 8*laneHi
#pragma unroll
        for (int r = 0; r < 8; ++r) {
            int m = r + 8 * laneHi;
            __builtin_nontemporal_store(acc[r],
                &obase[(size_t)(t0 + m) * rs + d0 + lanLo]);
        }

        // new carry = full h at row 15: lanes 16..31 hold M=15 in VGPR 7
        carry = __shfl(acc[7], 16 + lanLo, 32);
    }
}

extern "C" void kernel_launch(void* const* d_in, const int* in_sizes, int n_in,
                              void* d_out, int out_size, void* d_ws, size_t ws_size,
                              hipStream_t stream) {
    const float* x           = (const float*)d_in[0];
    const float* alpha_logit = (const float*)d_in[1];
    float* out = (float*)d_out;

    float* segf  = (float*)d_ws;                          // [B, SEGS, DCH]
    float* carry = segf + (size_t)BATCH * SEGS * DCH;     // [B, SEGS, DCH]

    {
        int threads = BATCH * SEGS * DCH;                 // 131072
        ema_seg_final<<<threads / 256, 256, 0, stream>>>(x, alpha_logit, segf);
    }
    {
        int threads = BATCH * DCH;                        // 8192
        ema_carry_scan<<<threads / 256, 256, 0, stream>>>(segf, alpha_logit, carry);
    }
    {
        int waves = BATCH * SEGS * NTILE;                 // 8192 waves
        ema_scan_wmma<<<waves / 8, 256, 0, stream>>>(x, alpha_logit, carry, out);
    }
}